// TritonSIRENDownsample1D_85109071937778
// MI455X (gfx1250) — compile-verified
//
#include <hip/hip_runtime.h>
#include <hip/hip_bf16.h>

typedef __attribute__((ext_vector_type(2))) float v2f;
typedef __attribute__((ext_vector_type(4))) float fx4;
typedef __attribute__((ext_vector_type(8))) float v8f;

#define OMEGA0 30.0f
#define CH    512
#define HID   32
#define KS    4
#define LIN   16384
#define LOUT  4096
#define BATCH 8

// ---------------------------------------------------------------------------
// Stage 1: SIREN MLP -> conv kernel, via V_WMMA_F32_16X16X4_F32.
// One wave (32 lanes), EXEC all ones (WMMA requirement).
//
// A (16x4 f32):  lanes 0-15 hold {K=0,K=1}, lanes 16-31 hold {K=2,K=3}; M=lane&15
// B (4x16 f32):  mirrored: vgpr0 = row(K=kh), vgpr1 = row(K=kh+1); N=lane&15
// C/D (16x16):   lanes 0-15 rows 0-7 in acc[0..7]; lanes 16-31 rows 8-15
// ---------------------------------------------------------------------------

static __device__ inline v8f wmma_f32_4(v2f a, v2f b, v8f c) {
  // 8 args: (neg_a, A, neg_b, B, c_mod, C, reuse_a, reuse_b)
  return __builtin_amdgcn_wmma_f32_16x16x4_f32(false, a, false, b, (short)0, c,
                                               false, false);
}

// One 16-column output tile of: out[m][n] = sum_j hin[m][j] * W[n*HID + j]
static __device__ inline v8f dense_tile(const float (*hin)[HID],
                                        const float* __restrict__ W,
                                        int n, int kh) {
  v8f acc = {};
  const int m = threadIdx.x & 15;
#pragma unroll
  for (int kk = 0; kk < HID; kk += 4) {
    v2f a, b;
    a.x = hin[m][kk + kh];
    a.y = hin[m][kk + kh + 1];
    b.x = W[n * HID + kk + kh];
    b.y = W[n * HID + kk + kh + 1];
    acc = wmma_f32_4(a, b, acc);
  }
  return acc;
}

__global__ __launch_bounds__(32)
void siren_build_kernel(const float* __restrict__ w0, const float* __restrict__ b0,
                        const float* __restrict__ w1, const float* __restrict__ b1,
                        const float* __restrict__ w2, const float* __restrict__ b2,
                        const float* __restrict__ w3, const float* __restrict__ b3,
                        float* __restrict__ kernelT /* [KS][CH] */) {
  __shared__ float hA[16][HID];
  __shared__ float hB[16][HID];
  const int lane  = threadIdx.x;
  const int nl    = lane & 15;
  const int kh    = (lane >> 4) << 1;  // K sub-offset: 0 (lanes 0-15) or 2
  const int rbase = (lane >> 4) << 3;  // D rows: 0 or 8

  // layer 0: hA[m][j] = sin(pos[m] * w0[j] + b0[j]); pad rows m>=KS with 0
  {
    float w = w0[lane], bb = b0[lane];
#pragma unroll
    for (int mm = 0; mm < 16; ++mm) {
      float p = -OMEGA0 + (2.0f * OMEGA0 / (KS - 1)) * (float)mm;
      hA[mm][lane] = (mm < KS) ? __sinf(p * w + bb) : 0.0f;
    }
  }
  __syncthreads();

  // layer 1: hA -> hB   ([16,32] @ [32,32]^T, 2 N-tiles x 8 K-steps)
#pragma unroll
  for (int t = 0; t < HID / 16; ++t) {
    int n = t * 16 + nl;
    v8f acc = dense_tile(hA, w1, n, kh);
    float bb = b1[n];
#pragma unroll
    for (int r = 0; r < 8; ++r) hB[rbase + r][n] = __sinf(acc[r] + bb);
  }
  __syncthreads();

  // layer 2: hB -> hA
#pragma unroll
  for (int t = 0; t < HID / 16; ++t) {
    int n = t * 16 + nl;
    v8f acc = dense_tile(hB, w2, n, kh);
    float bb = b2[n];
#pragma unroll
    for (int r = 0; r < 8; ++r) hA[rbase + r][n] = __sinf(acc[r] + bb);
  }
  __syncthreads();

  // output layer: kernelT[k][c] = sum_j hA[k][j]*w3[c*HID+j] + b3[c]
  // Stored transposed [KS][CH] so stage-2 loads are contiguous float4 per tap.
  for (int t = 0; t < CH / 16; ++t) {
    int n = t * 16 + nl;
    v8f acc = dense_tile(hA, w3, n, kh);
    if (lane < 16) {  // rows 0..KS-1 live in lanes 0-15, acc[0..KS-1]
      float bb = b3[n];
#pragma unroll
      for (int r = 0; r < KS; ++r) kernelT[r * CH + n] = acc[r] + bb;
    }
  }
}

// ---------------------------------------------------------------------------
// Stage 2: bandwidth-bound strided depthwise conv.
// One thread per output float4 (4 channels). Consecutive threads ->
// consecutive channels -> fully coalesced b128 traffic. x is streamed once
// (256 MB > 192 MB L2) so loads/stores are non-temporal; the 8 KB kernel
// table stays cached.
// ---------------------------------------------------------------------------
__global__ __launch_bounds__(256)
void downsample_kernel(const float* __restrict__ x,
                       const float* __restrict__ kernelT,
                       float* __restrict__ out) {
  const unsigned NC4 = CH / 4;  // 128 float4 per row
  unsigned tid = blockIdx.x * 256u + threadIdx.x;
  unsigned c4 = tid & (NC4 - 1);
  unsigned l  = (tid >> 7) & (LOUT - 1);
  unsigned b  = tid >> 19;

  const fx4* xv = (const fx4*)x;
  const fx4* kv = (const fx4*)kernelT;
  fx4*       ov = (fx4*)out;

  unsigned base = (b * LIN + l * KS) * NC4 + c4;

  fx4 acc = {0.0f, 0.0f, 0.0f, 0.0f};
#pragma unroll
  for (int k = 0; k < KS; ++k) {
    fx4 xk = __builtin_nontemporal_load(&xv[base + (unsigned)k * NC4]);
    fx4 wk = kv[(unsigned)k * NC4 + c4];
    acc.x = fmaf(xk.x, wk.x, acc.x);
    acc.y = fmaf(xk.y, wk.y, acc.y);
    acc.z = fmaf(xk.z, wk.z, acc.z);
    acc.w = fmaf(xk.w, wk.w, acc.w);
  }
  __builtin_nontemporal_store(acc, &ov[tid]);
}

extern "C" void kernel_launch(void* const* d_in, const int* in_sizes, int n_in,
                              void* d_out, int out_size, void* d_ws, size_t ws_size,
                              hipStream_t stream) {
  const float* x  = (const float*)d_in[0];
  const float* w0 = (const float*)d_in[1];
  const float* b0 = (const float*)d_in[2];
  const float* w1 = (const float*)d_in[3];
  const float* b1 = (const float*)d_in[4];
  const float* w2 = (const float*)d_in[5];
  const float* b2 = (const float*)d_in[6];
  const float* w3 = (const float*)d_in[7];
  const float* b3 = (const float*)d_in[8];
  float* kT  = (float*)d_ws;  // KS*CH floats = 8 KB scratch
  float* out = (float*)d_out;

  siren_build_kernel<<<1, 32, 0, stream>>>(w0, b0, w1, b1, w2, b2, w3, b3, kT);

  const unsigned total  = BATCH * LOUT * (CH / 4);  // 4,194,304 float4 outputs
  const unsigned blocks = total / 256;              // 16384
  downsample_kernel<<<blocks, 256, 0, stream>>>(x, kT, out);
}